// ShieldLoss_75986561401036
// MI455X (gfx1250) — compile-verified
//
#include <hip/hip_runtime.h>

// ---------------------------------------------------------------------------
// ShieldLoss: constr[b,r] = max(0, max_j preds[b,vp[4r+j]], max_j 1-preds[b,vm[4r+j]])
// out = 1 - mean(constr).
// Row-major LDS tile with padded stride 1026 dwords: 8B-aligned rows (legal for
// async b64 copies) AND conflict-free gathers (bank = (2*lane + v) % 64).
// Staged via CDNA5 global_load_async_to_lds_b64 (ASYNCcnt path).
// WMMA intentionally unused: the reduction operator is max, not +.
// ---------------------------------------------------------------------------

#define NUM_VARS      1000
#define ROWS          32                     // batch rows per block (== wave32 lanes)
#define VROW          1026                   // padded row stride in dwords
#define LDS_BYTES     (ROWS * VROW * 4)      // 131328 B  (< 320 KB WGP LDS)
#define THREADS       256
#define NWAVES        (THREADS / 32)
#define REQ_PER_BLOCK 1024

typedef int v2i __attribute__((ext_vector_type(2)));
typedef __attribute__((address_space(1))) v2i g_v2i;   // global (addrspace 1)
typedef __attribute__((address_space(3))) v2i l_v2i;   // LDS    (addrspace 3)
typedef __attribute__((address_space(1))) int g_i32;
typedef __attribute__((address_space(3))) int l_i32;

// 8B: global -> LDS, asynchronously (ASYNCcnt path).
__device__ __forceinline__ void stage_b64(const float* g, float* l) {
#if __has_builtin(__builtin_amdgcn_global_load_async_to_lds_b64)
    __builtin_amdgcn_global_load_async_to_lds_b64((g_v2i*)g, (l_v2i*)l, 0, 0);
#else
    unsigned lofs = (unsigned)(unsigned long long)(l_v2i*)l;
    asm volatile("global_load_async_to_lds_b64 %0, %1, off"
                 :: "v"(lofs), "v"(g)
                 : "memory");
#endif
}

__device__ __forceinline__ void wait_async0() {
#if __has_builtin(__builtin_amdgcn_s_wait_asynccnt)
    __builtin_amdgcn_s_wait_asynccnt(0);
#else
    asm volatile("s_wait_asynccnt 0x0" ::: "memory");
#endif
}

extern "C" __global__ void __launch_bounds__(THREADS)
shield_gather_kernel(const float* __restrict__ preds,
                     const int*   __restrict__ vplus,
                     const int*   __restrict__ vminus,
                     int num_req, int B,
                     float* __restrict__ partial)
{
    extern __shared__ float S[];   // tile: S[row*VROW + v]

    const int tid  = threadIdx.x;
    const int row0 = blockIdx.x * ROWS;

    // ---- Stage up to 32 rows x 1000 f32 into LDS with async b64 copies.
    // Each row of the tile is contiguous in global memory (coalesced), and the
    // runtime loop bound keeps the compiler from unrolling 32x.
    int rows_here = B - row0;
    if (rows_here > ROWS) rows_here = ROWS;
    for (int row = 0; row < rows_here; ++row) {
        const float* g = preds + (size_t)(row0 + row) * NUM_VARS;
        float*       l = &S[row * VROW];
        for (int c = tid; c < NUM_VARS / 2; c += THREADS)   // 500 b64 elems per row
            stage_b64(g + 2 * c, l + 2 * c);
    }
    wait_async0();
    __syncthreads();

    // ---- Gather + max. lane == batch row; r is wave-uniform => scalar index loads.
    const int  w        = tid >> 5;
    const int  lane     = tid & 31;
    const bool rowvalid = (row0 + lane) < B;
    const float* Sl     = &S[lane * VROW];   // per-lane row base (conflict-free banks)

    int rbeg = blockIdx.y * REQ_PER_BLOCK;
    int rend = rbeg + REQ_PER_BLOCK;
    if (rend > num_req) rend = num_req;

    float acc = 0.0f;
    for (int r = rbeg + w; r < rend; r += NWAVES) {
        const int  ru = __builtin_amdgcn_readfirstlane(r);       // force SGPR / s_load
        const int4 p  = *(const int4*)(vplus  + 4ll * ru);
        const int4 q  = *(const int4*)(vminus + 4ll * ru);
        float m = 0.0f;                                          // max(0, ...)
        m = fmaxf(m, Sl[p.x]);
        m = fmaxf(m, Sl[p.y]);
        m = fmaxf(m, Sl[p.z]);
        m = fmaxf(m, Sl[p.w]);
        m = fmaxf(m, 1.0f - Sl[q.x]);
        m = fmaxf(m, 1.0f - Sl[q.y]);
        m = fmaxf(m, 1.0f - Sl[q.z]);
        m = fmaxf(m, 1.0f - Sl[q.w]);
        acc += m;
    }
    if (!rowvalid) acc = 0.0f;

    // ---- Deterministic block reduction (wave32 shuffles, no atomics).
    #pragma unroll
    for (int off = 16; off > 0; off >>= 1)
        acc += __shfl_xor(acc, off, 32);

    __shared__ float wpart[NWAVES];
    if (lane == 0) wpart[w] = acc;
    __syncthreads();
    if (tid == 0) {
        float s = 0.0f;
        #pragma unroll
        for (int i = 0; i < NWAVES; ++i) s += wpart[i];
        partial[(size_t)blockIdx.y * gridDim.x + blockIdx.x] = s;
    }
}

extern "C" __global__ void __launch_bounds__(256)
shield_reduce_kernel(const float* __restrict__ partial, int n,
                     float* __restrict__ out, float inv_count)
{
    float acc = 0.0f;
    for (int i = threadIdx.x; i < n; i += 256) acc += partial[i];
    #pragma unroll
    for (int off = 16; off > 0; off >>= 1)
        acc += __shfl_xor(acc, off, 32);
    __shared__ float wpart[8];
    const int w = threadIdx.x >> 5;
    if ((threadIdx.x & 31) == 0) wpart[w] = acc;
    __syncthreads();
    if (threadIdx.x == 0) {
        float s = 0.0f;
        #pragma unroll
        for (int i = 0; i < 8; ++i) s += wpart[i];
        out[0] = 1.0f - s * inv_count;
    }
}

extern "C" void kernel_launch(void* const* d_in, const int* in_sizes, int n_in,
                              void* d_out, int out_size, void* d_ws, size_t ws_size,
                              hipStream_t stream) {
    const float* preds  = (const float*)d_in[0];
    const int*   vplus  = (const int*)d_in[1];
    const int*   vminus = (const int*)d_in[2];
    // in_sizes: [B*NUM_VARS, NNZ, NNZ, NNZ, NNZ, 1]; 4 literals per requirement.
    const int num_req = in_sizes[1] / 4;                 // 50000
    const int B       = in_sizes[0] / NUM_VARS;          // 256

    const int gx = (B + ROWS - 1) / ROWS;                         // 8
    const int gy = (num_req + REQ_PER_BLOCK - 1) / REQ_PER_BLOCK; // 49
    float* partial = (float*)d_ws;                       // gx*gy floats of scratch

    // Dynamic LDS of 131328 B exceeds the 64 KB default cap; raise it (idempotent).
    (void)hipFuncSetAttribute((const void*)shield_gather_kernel,
                              hipFuncAttributeMaxDynamicSharedMemorySize, LDS_BYTES);

    shield_gather_kernel<<<dim3(gx, gy), THREADS, LDS_BYTES, stream>>>(
        preds, vplus, vminus, num_req, B, partial);

    shield_reduce_kernel<<<1, 256, 0, stream>>>(
        partial, gx * gy, (float*)d_out, 1.0f / ((float)B * (float)num_req));
}